// VecCrossAttnBlockApi_28982439313694
// MI455X (gfx1250) — compile-verified
//
#include <hip/hip_runtime.h>
#include <math.h>

#define EPSV 1e-6f
#define NSLOPE 0.2f

typedef float v2f __attribute__((ext_vector_type(2)));
typedef float v8f __attribute__((ext_vector_type(8)));

__device__ __forceinline__ v8f wmma_f32_4(v2f a, v2f b, v8f c) {
  // D = A(16x4) * B(4x16) + C(16x16), fp32 WMMA on gfx1250
  return __builtin_amdgcn_wmma_f32_16x16x4_f32(false, a, false, b, (short)0, c, false, false);
}

// ---------------------------------------------------------------------------
// Generic WMMA GEMM:  Y[b] (O x M) = W (O x Kd, row-major) @ X[b] (Kd x M)
// Block tile 128(O) x 128(M); 8 waves in a 4x2 grid, each wave a 32x64 tile
// (2x4 WMMA tiles -> 8 WMMAs per K=4 step, 10 VMEM issues per 8 WMMAs).
// accum!=0 -> Y += result (C tile preloaded into accumulators).
// ---------------------------------------------------------------------------
__global__ __launch_bounds__(256) void gemm_wmma_k(
    const float* __restrict__ W, const float* __restrict__ X, float* Y,
    int O, int Kd, int M, long long xbs, long long ybs, int accum)
{
  const int tid  = threadIdx.x;
  const int wid  = tid >> 5;
  const int lane = tid & 31;
  const int half = lane >> 4;
  const int l16  = lane & 15;
  const int r0 = blockIdx.y * 128 + (wid >> 1) * 32;   // wave row tile (32 rows)
  const int c0 = blockIdx.x * 128 + (wid & 1) * 64;    // wave col tile (64 cols)
  const float* __restrict__ Xb = X + (long long)blockIdx.z * xbs;
  float* Yb = Y + (long long)blockIdx.z * ybs;

  v8f acc[2][4];
  if (accum) {
    #pragma unroll
    for (int i = 0; i < 2; ++i)
      #pragma unroll
      for (int jj = 0; jj < 4; ++jj) {
        const int rb = r0 + i * 16 + half * 8;
        const int cc = c0 + jj * 16 + l16;
        #pragma unroll
        for (int j = 0; j < 8; ++j)
          acc[i][jj][j] = Yb[(long long)(rb + j) * M + cc];
      }
  } else {
    #pragma unroll
    for (int i = 0; i < 2; ++i)
      #pragma unroll
      for (int jj = 0; jj < 4; ++jj)
        acc[i][jj] = (v8f){0.f,0.f,0.f,0.f,0.f,0.f,0.f,0.f};
  }

  for (int k = 0; k < Kd; k += 4) {
    const int kk = k + 2 * half;
    // A fragments: lane row = l16 (both halves), K pair = kk,kk+1 (contiguous)
    v2f a0 = *(const v2f*)(W + (long long)(r0 + l16) * Kd + kk);
    v2f a1 = *(const v2f*)(W + (long long)(r0 + 16 + l16) * Kd + kk);
    // B fragments: lane col = l16, K rows kk,kk+1 (stride M apart), 4 col tiles
    v2f bf[4];
    #pragma unroll
    for (int jj = 0; jj < 4; ++jj) {
      bf[jj].x = Xb[(long long)kk       * M + c0 + jj * 16 + l16];
      bf[jj].y = Xb[(long long)(kk + 1) * M + c0 + jj * 16 + l16];
    }
    #pragma unroll
    for (int jj = 0; jj < 4; ++jj) {
      acc[0][jj] = wmma_f32_4(a0, bf[jj], acc[0][jj]);
      acc[1][jj] = wmma_f32_4(a1, bf[jj], acc[1][jj]);
    }
  }

  #pragma unroll
  for (int i = 0; i < 2; ++i)
    #pragma unroll
    for (int jj = 0; jj < 4; ++jj) {
      const int rb = r0 + i * 16 + half * 8;
      const int cc = c0 + jj * 16 + l16;
      #pragma unroll
      for (int j = 0; j < 8; ++j)
        Yb[(long long)(rb + j) * M + cc] = acc[i][jj][j];
    }
}

// ---------------------------------------------------------------------------
// Flash-style vector-neuron attention, one (b,h): Q,K,V are (48 x N) slabs of
// the (B,C,3,N) buffers. Computes O = V * softmax_cols(K^T Q * scale) fused.
// Block = 8 waves; wave owns 16 queries. S^T tiles via WMMA; P^T tile reused
// as WMMA B-operand through lane-half shuffles (D-layout -> B-layout).
// ---------------------------------------------------------------------------
__global__ __launch_bounds__(256) void flash_attn_k(
    const float* __restrict__ Q, const float* __restrict__ Kp,
    const float* __restrict__ V, float* __restrict__ Out,
    int NQ, int NV, float scale)
{
  const int b = blockIdx.z, h = blockIdx.y;
  const float* Qm = Q   + ((long long)b * 128 + h * 16) * 3 * NQ;
  const float* Km = Kp  + ((long long)b * 128 + h * 16) * 3 * NV;
  const float* Vm = V   + ((long long)b * 128 + h * 16) * 3 * NV;
  float*       Om = Out + ((long long)b * 128 + h * 16) * 3 * NQ;

  const int tid  = threadIdx.x;
  const int wid  = tid >> 5;
  const int lane = tid & 31;
  const int half = lane >> 4;
  const int l16  = lane & 15;
  const int n0   = blockIdx.x * 128 + wid * 16;

  // Hoist pre-scaled Q B-fragments (constant over the key loop): 12 k-steps.
  v2f qf[12];
  #pragma unroll
  for (int ks = 0; ks < 12; ++ks) {
    const int rb = ks * 4 + 2 * half;
    v2f t;
    t.x = Qm[(long long)rb       * NQ + n0 + l16] * scale;
    t.y = Qm[(long long)(rb + 1) * NQ + n0 + l16] * scale;
    qf[ks] = t;
  }

  v8f o[3];
  #pragma unroll
  for (int ft = 0; ft < 3; ++ft) o[ft] = (v8f){0.f,0.f,0.f,0.f,0.f,0.f,0.f,0.f};
  float m_run = -3.0e38f, l_run = 0.f;

  for (int m0 = 0; m0 < NV; m0 += 16) {
    // S^T tile (rows = keys m, cols = queries n)
    v8f st = (v8f){0.f,0.f,0.f,0.f,0.f,0.f,0.f,0.f};
    #pragma unroll
    for (int ks = 0; ks < 12; ++ks) {
      const int rb = ks * 4 + 2 * half;
      v2f kf;
      kf.x = Km[(long long)rb       * NV + m0 + l16];
      kf.y = Km[(long long)(rb + 1) * NV + m0 + l16];
      st = wmma_f32_4(kf, qf[ks], st);
    }

    // Online softmax over key rows (8 regs in-lane + lane-half exchange).
    float tmax = st[0];
    #pragma unroll
    for (int j = 1; j < 8; ++j) tmax = fmaxf(tmax, st[j]);
    tmax = fmaxf(tmax, __shfl_xor(tmax, 16, 32));
    const float newm = fmaxf(m_run, tmax);
    const float sc = __expf(m_run - newm);

    float p[8];
    float tsum = 0.f;
    #pragma unroll
    for (int j = 0; j < 8; ++j) { p[j] = __expf(st[j] - newm); tsum += p[j]; }
    tsum += __shfl_xor(tsum, 16, 32);
    l_run = l_run * sc + tsum;
    m_run = newm;

    #pragma unroll
    for (int ft = 0; ft < 3; ++ft)
      #pragma unroll
      for (int j = 0; j < 8; ++j) o[ft][j] *= sc;

    // Cross-half copies of P^T rows (D-layout -> B-layout conversion).
    float sp[8];
    #pragma unroll
    for (int j = 0; j < 8; ++j) sp[j] = __shfl_xor(p[j], 16, 32);

    // O^T += V_tile(16f x 16m) * P^T(16m x 16n): 3 feature tiles x 4 k-steps.
    #pragma unroll
    for (int ft = 0; ft < 3; ++ft) {
      #pragma unroll
      for (int ks2 = 0; ks2 < 4; ++ks2) {
        const int k0 = ks2 * 4;
        v2f af = *(const v2f*)(Vm + (long long)(ft * 16 + l16) * NV + m0 + k0 + 2 * half);
        v2f bf;
        const int i0 = (ks2 & 1) * 4;
        if (ks2 < 2) { // rows 0..7 of P^T
          bf.x = half ? sp[i0 + 2] : p[i0];
          bf.y = half ? sp[i0 + 3] : p[i0 + 1];
        } else {       // rows 8..15 of P^T
          bf.x = half ? p[i0 + 2] : sp[i0];
          bf.y = half ? p[i0 + 3] : sp[i0 + 1];
        }
        o[ft] = wmma_f32_4(af, bf, o[ft]);
      }
    }
  }

  const float rl = 1.0f / l_run;
  #pragma unroll
  for (int ft = 0; ft < 3; ++ft)
    #pragma unroll
    for (int j = 0; j < 8; ++j) {
      const int r = ft * 16 + j + 8 * half;
      Om[(long long)r * NQ + n0 + l16] = o[ft][j] * rl;
    }
}

// ---------------------------------------------------------------------------
// vec_layernorm: X (B,128,3,N) -> Y. Norm over t, mean/var over C.
// Block (32 n's, 8 c-strides).
// ---------------------------------------------------------------------------
__global__ __launch_bounds__(256) void vec_layernorm_k(
    const float* __restrict__ X, const float* __restrict__ gamma,
    const float* __restrict__ beta, float* __restrict__ Y, int N)
{
  __shared__ float nrm_s[128][33];
  __shared__ float red1[8][32];
  __shared__ float red2[8][32];
  const int tx = threadIdx.x, ty = threadIdx.y;
  const int n = blockIdx.x * 32 + tx;
  const long long bb = (long long)blockIdx.y * 128 * 3 * N;

  float s1 = 0.f, s2 = 0.f;
  for (int c = ty; c < 128; c += 8) {
    const long long base = bb + (long long)c * 3 * N + n;
    float x0 = X[base], x1 = X[base + N], x2 = X[base + 2 * N];
    float nr = sqrtf(x0 * x0 + x1 * x1 + x2 * x2 + EPSV);
    nrm_s[c][tx] = nr;
    s1 += nr; s2 += nr * nr;
  }
  red1[ty][tx] = s1; red2[ty][tx] = s2;
  __syncthreads();
  if (ty == 0) {
    #pragma unroll
    for (int r = 1; r < 8; ++r) { s1 += red1[r][tx]; s2 += red2[r][tx]; }
    const float mu = s1 * (1.0f / 128.0f);
    red1[0][tx] = mu;
    red2[0][tx] = s2 * (1.0f / 128.0f) - mu * mu;
  }
  __syncthreads();
  const float mu = red1[0][tx];
  const float rstd = rsqrtf(red2[0][tx] + EPSV);
  for (int c = ty; c < 128; c += 8) {
    const float nr = nrm_s[c][tx];
    const float ln = (nr - mu) * rstd * gamma[c] + beta[c];
    const float f = ln / (nr + EPSV);
    const long long base = bb + (long long)c * 3 * N + n;
    Y[base]         = X[base] * f;
    Y[base + N]     = X[base + N] * f;
    Y[base + 2 * N] = X[base + 2 * N] * f;
  }
}

// ---------------------------------------------------------------------------
// Graph attention gather: per (b,n) compute 16 neighbor scores, softmax,
// and write weighted edge features: EDGE (B, 2C, 3, NQ):
//   rows 0..C-1   : sum_k a_k * v[idx_k] - center
//   rows C..2C-1  : center
// (uses the linearity of feat: out = We @ weighted-edge)
// ---------------------------------------------------------------------------
__global__ __launch_bounds__(128) void graph_gather_k(
    const float* __restrict__ XQ, const float* __restrict__ XV,
    const int* __restrict__ IDX, float* __restrict__ EDGE,
    int NQ, int NV, float scale)
{
  __shared__ int   idx_s[16];
  __shared__ float sarr[128][17];
  __shared__ float a_s[16];
  __shared__ float red[16];
  const int b = blockIdx.y, n = blockIdx.x, c = threadIdx.x;

  if (c < 16) idx_s[c] = IDX[((long long)b * NQ + n) * 16 + c];
  __syncthreads();

  const long long qb  = ((long long)b * 128 + c) * 3 * NQ;
  const long long vb0 = ((long long)b * 128 + c) * 3 * NV;
  const float qv0 = XQ[qb + n], qv1 = XQ[qb + NQ + n], qv2 = XQ[qb + 2 * NQ + n];

  #pragma unroll 4
  for (int k = 0; k < 16; ++k) {
    const long long vb = vb0 + idx_s[k];
    sarr[c][k] = qv0 * XV[vb] + qv1 * XV[vb + NV] + qv2 * XV[vb + 2 * NV];
  }
  __syncthreads();
  if (c < 16) {
    float s = 0.f;
    for (int j = 0; j < 128; ++j) s += sarr[j][c];
    red[c] = s * scale;
  }
  __syncthreads();
  if (c == 0) {
    float mx = red[0];
    #pragma unroll
    for (int k = 1; k < 16; ++k) mx = fmaxf(mx, red[k]);
    float sum = 0.f;
    #pragma unroll
    for (int k = 0; k < 16; ++k) { a_s[k] = __expf(red[k] - mx); sum += a_s[k]; }
    const float inv = 1.0f / sum;
    #pragma unroll
    for (int k = 0; k < 16; ++k) a_s[k] *= inv;
  }
  __syncthreads();

  float w0 = 0.f, w1 = 0.f, w2 = 0.f;
  #pragma unroll 4
  for (int k = 0; k < 16; ++k) {
    const long long vb = vb0 + idx_s[k];
    const float a = a_s[k];
    w0 += a * XV[vb]; w1 += a * XV[vb + NV]; w2 += a * XV[vb + 2 * NV];
  }
  const long long e0 = ((long long)b * 256 + c) * 3 * NQ;
  const long long e1 = ((long long)b * 256 + 128 + c) * 3 * NQ;
  EDGE[e0 + n]          = w0 - qv0;
  EDGE[e0 + NQ + n]     = w1 - qv1;
  EDGE[e0 + 2 * NQ + n] = w2 - qv2;
  EDGE[e1 + n]          = qv0;
  EDGE[e1 + NQ + n]     = qv1;
  EDGE[e1 + 2 * NQ + n] = qv2;
}

// ---------------------------------------------------------------------------
// vn_leaky (in place on H): per (b,ch,n) vector of 3.
// ---------------------------------------------------------------------------
__global__ __launch_bounds__(256) void vn_leaky_k(
    float* __restrict__ H, const float* __restrict__ D, int N, long long total)
{
  const long long i = (long long)blockIdx.x * blockDim.x + threadIdx.x;
  if (i >= total) return;
  const int n = (int)(i % N);
  const long long bc = i / N;
  const long long base = bc * 3 * N + n;
  const float h0 = H[base], h1 = H[base + N], h2 = H[base + 2 * N];
  const float d0 = D[base], d1 = D[base + N], d2 = D[base + 2 * N];
  const float dot = h0 * d0 + h1 * d1 + h2 * d2;
  const float dsq = d0 * d0 + d1 * d1 + d2 * d2;
  const float f = dot / (dsq + EPSV);
  const bool pos = (dot >= 0.f);
  const float x0 = pos ? h0 : (h0 - f * d0);
  const float x1 = pos ? h1 : (h1 - f * d1);
  const float x2 = pos ? h2 : (h2 - f * d2);
  H[base]         = NSLOPE * h0 + (1.0f - NSLOPE) * x0;
  H[base + N]     = NSLOPE * h1 + (1.0f - NSLOPE) * x1;
  H[base + 2 * N] = NSLOPE * h2 + (1.0f - NSLOPE) * x2;
}

__global__ __launch_bounds__(256) void copy_k(const float* __restrict__ src,
                                              float* __restrict__ dst, long long n)
{
  const long long i = (long long)blockIdx.x * blockDim.x + threadIdx.x;
  if (i < n) dst[i] = src[i];
}

// ---------------------------------------------------------------------------
extern "C" void kernel_launch(void* const* d_in, const int* in_sizes, int n_in,
                              void* d_out, int out_size, void* d_ws, size_t ws_size,
                              hipStream_t stream) {
  (void)in_sizes; (void)n_in; (void)out_size; (void)ws_size;
  const int Bn = 2, Cc = 128, N = 2048, H = 8, HIDc = 512;
  const int M = 3 * N; // 6144

  const float* q   = (const float*)d_in[0];
  const float* v   = (const float*)d_in[1];
  const int*   idxS = (const int*)d_in[4];
  const int*   idxC = (const int*)d_in[5];
  const float* g1 = (const float*)d_in[6],  *b1 = (const float*)d_in[7];
  const float* g2 = (const float*)d_in[8],  *b2 = (const float*)d_in[9];
  const float* gq = (const float*)d_in[10], *bq = (const float*)d_in[11];
  const float* gv = (const float*)d_in[12], *bv = (const float*)d_in[13];
  const float* Wq_s = (const float*)d_in[14];
  const float* Wk_s = (const float*)d_in[15];
  const float* Wv_s = (const float*)d_in[16];
  const float* Wo_s = (const float*)d_in[17];
  const float* We_s = (const float*)d_in[18];
  const float* Wm_s = (const float*)d_in[19];
  const float* Wq_c = (const float*)d_in[20];
  const float* Wk_c = (const float*)d_in[21];
  const float* Wv_c = (const float*)d_in[22];
  const float* Wo_c = (const float*)d_in[23];
  const float* We_c = (const float*)d_in[24];
  const float* Wm_c = (const float*)d_in[25];
  const float* W1   = (const float*)d_in[26];
  const float* Dact = (const float*)d_in[27];
  const float* W2   = (const float*)d_in[28];
  float* out = (float*)d_out;

  const long long sz1 = (long long)Bn * Cc * M; // 1,572,864 floats
  float* ws   = (float*)d_ws;
  float* nx   = ws;
  float* nvb  = nx  + sz1;
  float* qb   = nvb + sz1;
  float* kb   = qb  + sz1;
  float* vb   = kb  + sz1;
  float* attn = vb  + sz1;
  float* edge = attn + sz1;      // 2*sz1
  float* cat  = edge + 2 * sz1;  // 2*sz1
  float* hb   = cat  + 2 * sz1;  // 4*sz1
  float* db   = hb   + 4 * sz1;  // 4*sz1

  const float attn_scale  = 1.0f / sqrtf(48.0f);   // sqrt(3*Dh)
  const float graph_scale = 1.0f / sqrtf(384.0f);  // sqrt(3*C)

  const dim3 lnBlk(32, 8);
  const dim3 lnGrd(N / 32, Bn);
  const dim3 faGrd(N / 128, H, Bn);
  const dim3 ggGrd(N, Bn);

  auto gemm = [&](const float* W, const float* X, float* Y, int O, int Kd,
                  long long xbs, long long ybs, int acc) {
    dim3 g(M / 128, O / 128, Bn);
    gemm_wmma_k<<<g, 256, 0, stream>>>(W, X, Y, O, Kd, M, xbs, ybs, acc);
  };

  // out = q (residual accumulator)
  copy_k<<<(int)(sz1 / 256), 256, 0, stream>>>(q, out, sz1);

  // ---- Self block ----
  vec_layernorm_k<<<lnGrd, lnBlk, 0, stream>>>(q, g1, b1, nx, N);
  gemm(Wq_s, nx, qb, Cc, Cc, sz1, sz1, 0);
  gemm(Wk_s, nx, kb, Cc, Cc, sz1, sz1, 0);
  gemm(Wv_s, nx, vb, Cc, Cc, sz1, sz1, 0);
  flash_attn_k<<<faGrd, 256, 0, stream>>>(qb, kb, vb, attn, N, N, attn_scale);
  gemm(Wo_s, attn, cat, Cc, Cc, sz1, 2 * sz1, 0);                 // concat[0:C]
  graph_gather_k<<<ggGrd, 128, 0, stream>>>(nx, nx, idxS, edge, N, N, graph_scale);
  gemm(We_s, edge, cat + (long long)Cc * M, Cc, 2 * Cc, 2 * sz1, 2 * sz1, 0); // concat[C:2C]
  gemm(Wm_s, cat, out, Cc, 2 * Cc, 2 * sz1, sz1, 1);              // out += Wm_s@cat

  // ---- Cross block ----
  vec_layernorm_k<<<lnGrd, lnBlk, 0, stream>>>(out, gq, bq, nx, N);
  vec_layernorm_k<<<lnGrd, lnBlk, 0, stream>>>(v,   gv, bv, nvb, N);
  gemm(Wq_c, nx,  qb, Cc, Cc, sz1, sz1, 0);
  gemm(Wk_c, nvb, kb, Cc, Cc, sz1, sz1, 0);
  gemm(Wv_c, nvb, vb, Cc, Cc, sz1, sz1, 0);
  flash_attn_k<<<faGrd, 256, 0, stream>>>(qb, kb, vb, attn, N, N, attn_scale);
  gemm(Wo_c, attn, cat, Cc, Cc, sz1, 2 * sz1, 0);
  graph_gather_k<<<ggGrd, 128, 0, stream>>>(nx, nvb, idxC, edge, N, N, graph_scale);
  gemm(We_c, edge, cat + (long long)Cc * M, Cc, 2 * Cc, 2 * sz1, 2 * sz1, 0);
  gemm(Wm_c, cat, out, Cc, 2 * Cc, 2 * sz1, sz1, 1);

  // ---- MLP block ----
  vec_layernorm_k<<<lnGrd, lnBlk, 0, stream>>>(out, g2, b2, nx, N);
  gemm(W1,   nx, hb, HIDc, Cc,   sz1,     4 * sz1, 0);
  gemm(Dact, hb, db, HIDc, HIDc, 4 * sz1, 4 * sz1, 0);
  const long long lk_total = (long long)Bn * HIDc * N;
  vn_leaky_k<<<(int)((lk_total + 255) / 256), 256, 0, stream>>>(hb, db, N, lk_total);
  gemm(W2, hb, out, Cc, HIDc, 4 * sz1, sz1, 1);
}